// RandomPool_65627100283555
// MI455X (gfx1250) — compile-verified
//
#include <hip/hip_runtime.h>

// RandomPool gather for MI455X (gfx1250, wave32).
// out[b,c,i,j] = x[b,c, 2i + sel/2, 2j + sel%2],  sel = select_idx[b/T, i*56+j]
//
// Memory-bound: per thread (4 output pixels) load the 16-float source window
// (rows 2i,2i+1 x cols 8q..8q+7) as four b128 NT loads sharing ONE 64-bit
// address (immediate offsets 0/16/448/464 B), select with v_cndmask, write one
// b128 NT store. A tiny prep kernel precomputes clip = b / T so the hot kernel
// has no integer division.

typedef __attribute__((ext_vector_type(4))) float v4f;
typedef __attribute__((ext_vector_type(4))) int   v4i;

#define CCH    128     // channels
#define WW     112     // input width (floats); row stride = 448 B
#define SIDE   56      // output side
#define NPATCH 3136    // 56*56
#define PLANE  12544   // 112*112
#define Q_PER_ROW 14   // 56 / 4 float4 groups per output row

// ---- prep: clipTab[b] = b / T  (removes runtime division from hot kernel) ----
__global__ void randompool_clip_kernel(const int* __restrict__ Tptr,
                                       int* __restrict__ clipTab, int B) {
    int t = threadIdx.x;
    if (t < B) clipTab[t] = t / Tptr[0];
}

// ---- main gather ----
__global__ __launch_bounds__(256) void randompool_gather_kernel(
    const float* __restrict__ x,
    const int*   __restrict__ sel_idx,
    const int*   __restrict__ clipTab,
    float*       __restrict__ out,
    int total4)                       // number of float4 output groups
{
    int tid = blockIdx.x * blockDim.x + threadIdx.x;
    if (tid >= total4) return;

    // tid -> (plane r2 = b*C + c, output row i, quad q); out addr == out + 4*tid
    int q  = tid % Q_PER_ROW;
    int r1 = tid / Q_PER_ROW;
    int i  = r1 % SIDE;
    int r2 = r1 / SIDE;               // plane index: b*128 + c
    int b  = r2 >> 7;                 // / CCH
    int clip = clipTab[b];            // L2-hot, replaces b / T

    // One 64-bit base covers all four x loads via immediate offsets.
    const float* __restrict__ xbase =
        x + (size_t)r2 * PLANE + (size_t)(2 * i) * WW + 8 * q;

    // Source window: rows 2i, 2i+1, cols 8q..8q+7 (covers both dc parities).
    v4f lo0 = __builtin_nontemporal_load((const v4f*)(xbase));          // +0
    v4f hi0 = __builtin_nontemporal_load((const v4f*)(xbase + 4));      // +16B
    v4f lo1 = __builtin_nontemporal_load((const v4f*)(xbase + WW));     // +448B
    v4f hi1 = __builtin_nontemporal_load((const v4f*)(xbase + WW + 4)); // +464B

    // 4 indices, 16B-aligned (clip*3136, i*56, 4q all multiples of 4 ints).
    const v4i s4 = *(const v4i*)(sel_idx + clip * NPATCH + i * SIDE + 4 * q);

    v4f v;
    {   // j = 4q+0 -> cols {0,1} of window
        int s = s4.x;
        float a = (s & 1) ? lo0.y : lo0.x;   // row 2i
        float c = (s & 1) ? lo1.y : lo1.x;   // row 2i+1
        v.x = (s & 2) ? c : a;
    }
    {   // j = 4q+1 -> cols {2,3}
        int s = s4.y;
        float a = (s & 1) ? lo0.w : lo0.z;
        float c = (s & 1) ? lo1.w : lo1.z;
        v.y = (s & 2) ? c : a;
    }
    {   // j = 4q+2 -> cols {4,5}
        int s = s4.z;
        float a = (s & 1) ? hi0.y : hi0.x;
        float c = (s & 1) ? hi1.y : hi1.x;
        v.z = (s & 2) ? c : a;
    }
    {   // j = 4q+3 -> cols {6,7}
        int s = s4.w;
        float a = (s & 1) ? hi0.w : hi0.z;
        float c = (s & 1) ? hi1.w : hi1.z;
        v.w = (s & 2) ? c : a;
    }

    // Streamed output: non-temporal 128-bit store, fully coalesced.
    __builtin_nontemporal_store(v, (v4f*)(out + (size_t)tid * 4));
}

extern "C" void kernel_launch(void* const* d_in, const int* in_sizes, int n_in,
                              void* d_out, int out_size, void* d_ws, size_t ws_size,
                              hipStream_t stream) {
    (void)ws_size; (void)n_in; (void)in_sizes;

    const float* x   = (const float*)d_in[0];  // (32,128,112,112) f32
    const int*   sel = (const int*)  d_in[1];  // (4,3136) i32
    const int*   T   = (const int*)  d_in[2];  // scalar T (=8)
    float*       out = (float*)d_out;          // (32,128,56,56) f32
    int*         clipTab = (int*)d_ws;         // B ints of scratch

    int nplanes = out_size / NPATCH;           // B*C = 4096
    int B       = nplanes / CCH;               // 32
    int total4  = out_size / 4;                // float4 groups

    randompool_clip_kernel<<<1, 64, 0, stream>>>(T, clipTab, B);

    int block = 256;                           // 8 wave32 per workgroup
    int grid  = (total4 + block - 1) / block;
    randompool_gather_kernel<<<grid, block, 0, stream>>>(x, sel, clipTab, out, total4);
}